// DyGFormer_80333068304565
// MI455X (gfx1250) — compile-verified
//
#include <hip/hip_runtime.h>

typedef __attribute__((ext_vector_type(2))) float v2f;
typedef __attribute__((ext_vector_type(8))) float v8f;

#define NPOS  512
#define FDIM  64
#define HBINS 1024

__global__ __launch_bounds__(256) void dygformer_fused_kernel(
    const int* __restrict__ src_ids, const int* __restrict__ dst_ids,
    const float* __restrict__ W1, const float* __restrict__ b1,
    const float* __restrict__ W2, const float* __restrict__ b2,
    float* __restrict__ out, int B)
{
    __shared__ int   histS[HBINS];
    __shared__ int   histD[HBINS];
    __shared__ float fc0[2][NPOS];   // first freq channel per encode
    __shared__ float fc1[2][NPOS];   // second freq channel per encode
    __shared__ float W1s[FDIM], b1s[FDIM], b2s[FDIM];
    __shared__ float W2s[FDIM * FDIM];

    const int tid = threadIdx.x;
    const int b   = blockIdx.x;
    const long long rowbase = (long long)b * NPOS;

    // ---- Phase 0: zero histograms, stage weights in LDS ----
    for (int i = tid; i < HBINS; i += 256) { histS[i] = 0; histD[i] = 0; }
    for (int i = tid; i < FDIM;  i += 256) { W1s[i] = W1[i]; b1s[i] = b1[i]; b2s[i] = b2[i]; }
    for (int i = tid; i < FDIM * FDIM; i += 256) W2s[i] = W2[i];
    __syncthreads();

    // ---- Phase 1: per-row histograms of src and dst ids ----
    for (int i = tid; i < NPOS; i += 256) {
        int s = src_ids[rowbase + i];
        int d = dst_ids[rowbase + i];
        if ((unsigned)s < (unsigned)HBINS) atomicAdd(&histS[s], 1);
        if ((unsigned)d < (unsigned)HBINS) atomicAdd(&histD[d], 1);
    }
    __syncthreads();

    // ---- Phase 2: frequency lookup (padded / OOB ids -> 0) ----
    for (int i = tid; i < NPOS; i += 256) {
        int s = src_ids[rowbase + i];
        int d = dst_ids[rowbase + i];
        float s0 = 0.f, s1 = 0.f, d0 = 0.f, d1 = 0.f;
        if ((unsigned)s < (unsigned)HBINS) { s0 = (float)histS[s]; s1 = (float)histD[s]; }
        if ((unsigned)d < (unsigned)HBINS) { d0 = (float)histD[d]; d1 = (float)histS[d]; }
        fc0[0][i] = s0; fc1[0][i] = s1;   // encode 0: source_freq = [src_in_src, src_in_dst]
        fc0[1][i] = d0; fc1[1][i] = d1;   // encode 1: dst_freq    = [dst_in_dst, dst_in_src]
    }
    __syncthreads();

    // ---- Phase 3: GEMM H @ W2 via V_WMMA_F32_16X16X4_F32 ----
    // H[m,k] = relu(c0[m]*W1[k]+b1[k]) + relu(c1[m]*W1[k]+b1[k]), built on the fly.
    const int lane = tid & 31;
    const int wave = tid >> 5;
    const int lrow = lane & 15;
    const int koff = (lane >> 4) << 1;   // A/B frag: upper half-wave holds K+2,K+3
    const int moff = (lane >> 4) << 3;   // C/D frag: upper half-wave holds M+8

    // 256 tiles: e(2) x mt(32) x nt(4); 8 waves -> uniform 32 tiles per wave.
    for (int t = wave; t < 256; t += 8) {
        const int e  = t >> 7;
        const int mt = (t >> 2) & 31;
        const int nt = t & 3;
        const int m  = mt * 16 + lrow;
        const int n  = nt * 16 + lrow;

        const float c0 = fc0[e][m];
        const float c1 = fc1[e][m];

        v8f acc = {};
        #pragma unroll
        for (int ks = 0; ks < 16; ++ks) {
            const int klo = ks * 4 + koff;
            const float wlo = W1s[klo],     blo = b1s[klo];
            const float whi = W1s[klo + 1], bhi = b1s[klo + 1];

            v2f a, bb;
            a.x = fmaxf(c0 * wlo + blo, 0.f) + fmaxf(c1 * wlo + blo, 0.f);
            a.y = fmaxf(c0 * whi + bhi, 0.f) + fmaxf(c1 * whi + bhi, 0.f);
            bb.x = W2s[klo * FDIM + n];
            bb.y = W2s[(klo + 1) * FDIM + n];

            // 8 args: (neg_a, A, neg_b, B, c_mod, C, reuse_a, reuse_b)
            acc = __builtin_amdgcn_wmma_f32_16x16x4_f32(
                false, a, false, bb, (short)0, acc, false, false);
        }

        const float bias = 2.0f * b2s[n];   // b2 added once per k-channel, summed over 2
        float* obase = out + (long long)e * ((long long)B * NPOS * FDIM);
        #pragma unroll
        for (int r = 0; r < 8; ++r) {
            const int mo = mt * 16 + r + moff;
            obase[(rowbase + mo) * (long long)FDIM + n] = acc[r] + bias;
        }
    }
}

extern "C" void kernel_launch(void* const* d_in, const int* in_sizes, int n_in,
                              void* d_out, int out_size, void* d_ws, size_t ws_size,
                              hipStream_t stream) {
    const int*   src_ids = (const int*)d_in[0];
    const int*   dst_ids = (const int*)d_in[1];
    const float* W1      = (const float*)d_in[2];
    const float* b1      = (const float*)d_in[3];
    const float* W2      = (const float*)d_in[4];
    const float* b2      = (const float*)d_in[5];
    float* out = (float*)d_out;

    const int B = in_sizes[0] / NPOS;   // 1024

    dygformer_fused_kernel<<<B, 256, 0, stream>>>(
        src_ids, dst_ids, W1, b1, W2, b2, out, B);
}